// ConvOverTimeLayer_53798760350164
// MI455X (gfx1250) — compile-verified
//
#include <hip/hip_runtime.h>

// ---------------------------------------------------------------------------
// out[b,0,c,hw] = sum_t x[b,t,c,hw] * W[c,t] + bias[c]
// x: [16,10,1024,14,14] f32, W: [1024,10], bias: [1024]
//
// HBM-bound (AI ~0.45 FLOP/B, ~141 MB traffic -> ~6us at 23.3 TB/s). The K=10
// contraction runs on V_WMMA_F32_16X16X4_F32 (3 chained WMMAs, K padded to 12).
//
// Operand mapping: A (16x4, MxK) = W[c,t] broadcast across all 16 rows;
// B (4x16, KxN) = x for 16 spatial positions (N = lane&15). Every row of D
// equals out[N], so D VGPR0 holds the result lane-indexed by spatial position:
// zero shuffles, zero LDS, one coalesced 64B store per wave.
//
// K-padding: t >= 10 handled by CLAMPING both load indices (in-bounds re-read
// of t=9, unconditional loads, no exec-masked branches) and zeroing the
// already-loaded weight with a cndmask select.
// ---------------------------------------------------------------------------

typedef __attribute__((ext_vector_type(2))) float v2f;
typedef __attribute__((ext_vector_type(8))) float v8f;

#define N_MAPS 1024
#define T_IN   10
#define HW     196                 // 14*14
#define NTOT   (16 * HW)           // 3136 (b,hw) positions per channel
#define TILES_PER_C (NTOT / 16)    // 196 tiles of 16 positions (exact)
#define WAVES_PER_BLOCK 8          // 256 threads, wave32

__global__ __launch_bounds__(256)
void conv_over_time_wmma_kernel(const float* __restrict__ x,
                                const float* __restrict__ W,
                                const float* __restrict__ bias,
                                float* __restrict__ out)
{
    const int lane = threadIdx.x & 31;
    const int wid  = blockIdx.x * WAVES_PER_BLOCK + (threadIdx.x >> 5);
    const int c    = wid / TILES_PER_C;          // channel (uniform per wave)
    const int tile = wid - c * TILES_PER_C;      // which 16-position tile

    const int half = lane >> 4;                  // K half: lanes 16-31 do K=2,3
    const int nidx = lane & 15;                  // B-matrix column (N = spatial)
    const int n    = tile * 16 + nidx;           // flattened (b, hw)
    const int b    = n / HW;
    const int hw   = n - b * HW;

    // x[b,t,c,hw] flat offset = ((b*T_IN + t)*N_MAPS + c)*HW + hw
    const long long t_stride = (long long)N_MAPS * HW;
    const long long base     = ((long long)b * T_IN * N_MAPS + (long long)c) * HW + hw;

    const float* __restrict__ wrow = W + c * T_IN;

    v8f acc = (v8f)0.0f;

    // B layout (ISA 7.12.2 K-striping): VGPR0 = row K=2*half, VGPR1 = K=2*half+1.
    // A layout: lane m holds row m; broadcast weights identical for all m.
    #pragma unroll
    for (int kk = 0; kk < 3; ++kk) {
        const int t0  = 4 * kk + 2 * half;
        const int t1  = t0 + 1;
        const int t0c = (t0 < T_IN) ? t0 : (T_IN - 1);  // clamped, safe index
        const int t1c = (t1 < T_IN) ? t1 : (T_IN - 1);

        v2f xb, wa;
        xb.x = x[base + (long long)t0c * t_stride];     // unconditional loads
        xb.y = x[base + (long long)t1c * t_stride];

        const float w0 = wrow[t0c];                     // unconditional loads
        const float w1 = wrow[t1c];
        wa.x = (t0 < T_IN) ? w0 : 0.0f;                 // cndmask, not branch
        wa.y = (t1 < T_IN) ? w1 : 0.0f;

        // 8 args: (neg_a, A, neg_b, B, c_mod, C, reuse_a, reuse_b)
        acc = __builtin_amdgcn_wmma_f32_16x16x4_f32(
                  false, wa, false, xb, (short)0, acc, false, false);
    }

    // All D rows identical -> VGPR0 holds out[N = lane&15] in every lane.
    if (lane < 16) {
        // out[b,0,c,hw] : one coalesced 64B store per wave
        out[(long long)b * (N_MAPS * HW) + (long long)c * HW + hw]
            = acc[0] + bias[c];
    }
}

extern "C" void kernel_launch(void* const* d_in, const int* in_sizes, int n_in,
                              void* d_out, int out_size, void* d_ws, size_t ws_size,
                              hipStream_t stream) {
    const float* x  = (const float*)d_in[0];
    const float* W  = (const float*)d_in[1];
    const float* bb = (const float*)d_in[2];
    float* out      = (float*)d_out;

    const int total_waves = N_MAPS * TILES_PER_C;          // 200,704
    dim3 grid(total_waves / WAVES_PER_BLOCK);              // 25,088 blocks
    dim3 block(32 * WAVES_PER_BLOCK);                      // 256 threads

    conv_over_time_wmma_kernel<<<grid, block, 0, stream>>>(x, W, bb, out);
}